// QueryAndGroup_14714557956474
// MI455X (gfx1250) — compile-verified
//
#include <hip/hip_runtime.h>

// ---------------------------------------------------------------------------
// QueryAndGroup (PointNet++ ball query + grouping) for MI455X / gfx1250.
//
// Inputs (setup_inputs order):
//   d_in[0] xyz      : (B=8, N=8192, 3)  f32
//   d_in[1] new_xyz  : (B=8, M=2048, 3)  f32
//   d_in[2] features : (B=8, C=128, N)   f32
// Output:
//   d_out : (B, 3+C=131, M, NS=32) f32
// Workspace:
//   d_ws  : int idx[B][M][NS]  (2 MB)
//
// Phase 1 uses V_WMMA_F32_16X16X4_F32 with the radius test folded into C:
//   A row m  = (-2qx, -2qy, -2qz, 1)          (16x4)
//   B col n  = ( px ,  py ,  pz , |p|^2)      (4x16)
//   C[m][n]  = |q_m|^2 - r^2                  (row-constant, built once)
//   D[m][n]  = dist^2(q_m, p_n) - r^2 ; in-ball iff D < 0
// ---------------------------------------------------------------------------

typedef float v2f __attribute__((ext_vector_type(2)));
typedef float v8f __attribute__((ext_vector_type(8)));

#define BATCH 8
#define NPTS  8192
#define NQ    2048
#define NCH   128
#define NS    32
#define R2    0.04f

__global__ __launch_bounds__(128) void ball_query_wmma_kernel(
    const float* __restrict__ xyz,      // (B, N, 3)
    const float* __restrict__ new_xyz,  // (B, M, 3)
    int* __restrict__ idx)              // (B, M, NS)
{
    const int lane = threadIdx.x & 31;
    const int wave = threadIdx.x >> 5;
    const int tile = blockIdx.x * 4 + wave;   // [0, B*M/16)
    const int b    = tile >> 7;               // tile / (M/16=128)
    const int m0   = (tile & 127) << 4;       // first query of this 16-row tile
    const int row  = lane & 15;               // A-matrix row owned by this lane

    // ---- A matrix (16x4 f32): VGPR0 = K0(l0-15)/K2(l16-31), VGPR1 = K1/K3
    const float* q = new_xyz + ((size_t)b * NQ + (m0 + row)) * 3;
    const float qx = q[0], qy = q[1], qz = q[2];
    v2f a;
    if (lane < 16) { a.x = -2.0f * qx; a.y = -2.0f * qy; }
    else           { a.x = -2.0f * qz; a.y = 1.0f;       }

    // ---- C accumulator: C[v] at lane L = |q_{v+8*(L>=16)}|^2 - r^2.
    // Built ONCE before the scan loop (shuffles hoisted out of the hot path).
    const float qn = qx * qx + qy * qy + qz * qz - R2;  // row (lane&15) value
    v8f cinit;
    #pragma unroll
    for (int v = 0; v < 8; ++v) {
        const float clo = __shfl(qn, v, 32);       // row v
        const float chi = __shfl(qn, v + 8, 32);   // row v+8
        cinit[v] = (lane < 16) ? clo : chi;
    }

    int  cnt      = 0;   // valid neighbors found for row `row` (lanes 0-15)
    int  firstIdx = 0;
    int* myout    = idx + ((size_t)b * NQ + (m0 + row)) * NS;

    for (int base = 0; base < NPTS; base += 16) {
        // ---- B matrix (4x16 f32): column = lane&15, mirrored striping of A
        const float* p  = xyz + ((size_t)b * NPTS + (base + (lane & 15))) * 3;
        const float  px = p[0], py = p[1], pz = p[2];
        v2f bb;
        if (lane < 16) { bb.x = px; bb.y = py; }
        else           { bb.x = pz; bb.y = px * px + py * py + pz * pz; }

        // D = A x B + C : D[m][n] = dist^2 - r^2
        v8f d = __builtin_amdgcn_wmma_f32_16x16x4_f32(
            /*neg_a=*/false, a, /*neg_b=*/false, bb,
            /*c_mod=*/(short)0, cinit, /*reuse_a=*/false, /*reuse_b=*/false);

        // ---- per-VGPR ballots vs inline 0. D layout: VGPR v, lane L ->
        // row v+8*(L>=16), col L&15: ballot bits [15:0] = row v, [31:16] = v+8.
        unsigned mask16 = 0;
        #pragma unroll
        for (int v = 0; v < 8; ++v) {
            const unsigned bal = (unsigned)__ballot(d[v] < 0.0f);
            if ((row & 7) == v)
                mask16 = (row < 8) ? (bal & 0xFFFFu) : (bal >> 16);
        }

        // ---- lanes 0-15 serially append ascending in-radius indices
        if (lane < 16 && cnt < NS) {
            while (mask16 && cnt < NS) {
                const int bit = __builtin_ctz(mask16);
                mask16 &= mask16 - 1;
                const int pi = base + bit;
                if (cnt == 0) firstIdx = pi;
                myout[cnt++] = pi;
            }
        }

        // ---- uniform early exit once every row has NS hits
        const unsigned notdone = (unsigned)__ballot(lane < 16 && cnt < NS);
        if (notdone == 0) break;
    }

    // ---- pad: remaining slots get first valid index (or 0 if none)
    if (lane < 16) {
        const int fill = (cnt > 0) ? firstIdx : 0;
        for (int j = cnt; j < NS; ++j) myout[j] = fill;
    }
}

// ---------------------------------------------------------------------------
// Gather/group: out[b][c][m][k] = (c<3) ? xyz[b][n][c] - new_xyz[b][m][c]
//                                       : features[b][c-3][n],  n = idx[b][m][k]
// Coalesced streaming writes (non-temporal), cached gathers.
// ---------------------------------------------------------------------------
__global__ __launch_bounds__(256) void group_gather_kernel(
    const float* __restrict__ xyz,
    const float* __restrict__ new_xyz,
    const float* __restrict__ feat,
    const int*   __restrict__ idx,
    float*       __restrict__ out,
    long total)
{
    const long i = (long)blockIdx.x * blockDim.x + threadIdx.x;
    if (i >= total) return;

    const int  k  = (int)(i & (NS - 1));
    long       t  = i >> 5;                 // (b*131 + c)*M + m
    const int  m  = (int)(t & (NQ - 1));
    t >>= 11;                               // b*131 + c
    const int  c  = (int)(t % 131);
    const int  b  = (int)(t / 131);

    const int n = idx[((long)b * NQ + m) * NS + k];

    float val;
    if (c < 3) {
        val = xyz[((long)b * NPTS + n) * 3 + c]
            - new_xyz[((long)b * NQ + m) * 3 + c];
    } else {
        val = feat[((long)b * NCH + (c - 3)) * NPTS + n];
    }
    __builtin_nontemporal_store(val, &out[i]);
}

extern "C" void kernel_launch(void* const* d_in, const int* in_sizes, int n_in,
                              void* d_out, int out_size, void* d_ws, size_t ws_size,
                              hipStream_t stream) {
    const float* xyz     = (const float*)d_in[0];
    const float* new_xyz = (const float*)d_in[1];
    const float* feat    = (const float*)d_in[2];
    float*       out     = (float*)d_out;
    int*         idx     = (int*)d_ws;   // B*M*NS ints = 2 MB

    // Phase 1: ball query. 1024 16-query tiles, 4 wave32 tiles per block.
    const int tiles  = BATCH * (NQ / 16);     // 1024
    ball_query_wmma_kernel<<<tiles / 4, 128, 0, stream>>>(xyz, new_xyz, idx);

    // Phase 2: gather/group. One thread per output element.
    const long total = (long)BATCH * (3 + NCH) * NQ * NS;  // 68,681,728
    const long blocks = (total + 255) / 256;
    group_gather_kernel<<<(int)blocks, 256, 0, stream>>>(xyz, new_xyz, feat, idx, out, total);
}